// SessionGraph_78718160601343
// MI455X (gfx1250) — compile-verified
//
#include <hip/hip_runtime.h>

// ---------------------------------------------------------------------------
// SessionGraph on MI455X (gfx1250, wave32, WMMA).
//
//   make_ids      : u_ids = nu[flat], i_ids = ni[flat], s_ids = ns[flat]
//   pack_whh      : repack Whh (2,200,50) into WMMA B-fragment order
//   gemm_wmma     : pre = gather(table, ids) @ Wih_catT (100 -> 400) + b
//   lstm_cell_wmma: biLSTM cell (content T=2 or neighbor T=k), recurrence
//                   h @ WhhT on the WMMA pipe, time-mean output
//   attention     : leaky-relu scores, softmax over 4, weighted sum, pad mask
// ---------------------------------------------------------------------------

#define D_EMB 100   // embed dim
#define G4H   400   // 2 dirs * 4 gates * H
#define H_DIM 50
#define XLDS_STRIDE 104   // floats per LDS row (100 + pad), 16B-aligned rows

#define KT 13             // K tiles for recurrence: 52 >= H_DIM, steps of 4
#define CT 13             // col tiles for recurrence: 208 >= 200 gates
#define HS_STRIDE 56      // h LDS row stride (52 + pad)
#define GS_STRIDE 208     // gate LDS row stride

typedef __attribute__((ext_vector_type(2))) float v2f;
typedef __attribute__((ext_vector_type(8))) float v8f;

__device__ __forceinline__ v8f wmma_f32(v2f a, v2f b, v8f c) {
    return __builtin_amdgcn_wmma_f32_16x16x4_f32(false, a, false, b, (short)0, c, false, false);
}
__device__ __forceinline__ float sigf(float x) { return 1.0f / (1.0f + __expf(-x)); }

// ---------------------------------------------------------------------------
// Gather + GEMM: out[M x 400] = table[ids[m]] (M x 100) @ W^T (100 x 400) + bias
// ids == nullptr -> identity gather. 16 rows per workgroup; 4 waves split the
// 25 column tiles in pairs with two independent WMMA accumulator chains.
// ---------------------------------------------------------------------------
__global__ __launch_bounds__(128) void gemm_gather_wmma(
    const float* __restrict__ table, const int* __restrict__ ids,
    const float* __restrict__ W, const float* __restrict__ bias,
    float* __restrict__ out, int M)
{
    __shared__ float Xs[16 * XLDS_STRIDE];             // 16 gathered rows x 100
    __shared__ float Bs[4][2 * 16 * XLDS_STRIDE];      // per-wave: 2 B tiles

    const int tid  = threadIdx.x;
    const int wave = tid >> 5;
    const int lane = tid & 31;
    const int rowBase = blockIdx.x << 4;

    for (int e = tid; e < 16 * 25; e += 128) {
        const int r = e / 25, q = e - r * 25;
        const int src = ids ? ids[rowBase + r] : (rowBase + r);
        *(float4*)&Xs[r * XLDS_STRIDE + (q << 2)] =
            *(const float4*)&table[(size_t)src * D_EMB + (q << 2)];
    }
    __syncthreads();

    const int lh   = lane & 15;
    const int kofs = (lane >> 4) << 1;   // 0 or 2
    const int rAdd = (lane >> 4) << 3;   // 0 or 8

    v2f a[25];
#pragma unroll
    for (int ks = 0; ks < 25; ++ks)
        a[ks] = *(const v2f*)&Xs[lh * XLDS_STRIDE + (ks << 2) + kofs];

    float* Bw = &Bs[wave][0];

    for (int base = 0; base < 25; base += 8) {
        const int ct0 = base + wave;
        const int ct1 = base + wave + 4;
        if (ct0 >= 25) break;
        const bool dual = (ct1 < 25);

        {
            const int colBase0 = ct0 << 4;
            for (int e = lane; e < 16 * 25; e += 32) {
                const int n = e / 25, q = e - n * 25;
                *(float4*)&Bw[n * XLDS_STRIDE + (q << 2)] =
                    *(const float4*)&W[(size_t)(colBase0 + n) * D_EMB + (q << 2)];
            }
            if (dual) {
                const int colBase1 = ct1 << 4;
                for (int e = lane; e < 16 * 25; e += 32) {
                    const int n = e / 25, q = e - n * 25;
                    *(float4*)&Bw[16 * XLDS_STRIDE + n * XLDS_STRIDE + (q << 2)] =
                        *(const float4*)&W[(size_t)(colBase1 + n) * D_EMB + (q << 2)];
                }
            }
        }
        if (base + 8 + wave < 25)
            __builtin_prefetch(&W[(size_t)((base + 8 + wave) << 4) * D_EMB], 0, 0);

        __builtin_amdgcn_wave_barrier();

        v8f acc0 = {}, acc1 = {};
        if (dual) {
#pragma unroll
            for (int ks = 0; ks < 25; ++ks) {
                const int k0 = (ks << 2) + kofs;
                v2f b0 = *(const v2f*)&Bw[lh * XLDS_STRIDE + k0];
                v2f b1 = *(const v2f*)&Bw[(16 + lh) * XLDS_STRIDE + k0];
                acc0 = wmma_f32(a[ks], b0, acc0);
                acc1 = wmma_f32(a[ks], b1, acc1);
            }
        } else {
#pragma unroll
            for (int ks = 0; ks < 25; ++ks) {
                const int k0 = (ks << 2) + kofs;
                v2f b0 = *(const v2f*)&Bw[lh * XLDS_STRIDE + k0];
                acc0 = wmma_f32(a[ks], b0, acc0);
            }
        }
        __builtin_amdgcn_wave_barrier();

        {
            const int colBase = ct0 << 4;
            const float bv = bias[colBase + lh];
#pragma unroll
            for (int r = 0; r < 8; ++r)
                out[(size_t)(rowBase + rAdd + r) * G4H + colBase + lh] = acc0[r] + bv;
        }
        if (dual) {
            const int colBase = ct1 << 4;
            const float bv = bias[colBase + lh];
#pragma unroll
            for (int r = 0; r < 8; ++r)
                out[(size_t)(rowBase + rAdd + r) * G4H + colBase + lh] = acc1[r] + bv;
        }
    }
}

// ---------------------------------------------------------------------------
// Repack Whh (2,200,50) into B-fragment order: [d][ct][ks][lane] x v2f, where
//   col = ct*16 + (lane&15), k = ks*4 + ((lane>>4)*2), zero-padded to 208x52.
// One WMMA B operand then = one coalesced 256B load (hot in WGP$/L2).
// ---------------------------------------------------------------------------
__global__ void pack_whh(const float* __restrict__ Whh, float* __restrict__ outP)
{
    const int idx = blockIdx.x * blockDim.x + threadIdx.x;
    if (idx >= 2 * CT * KT * 32) return;
    const int lane = idx & 31;
    int t = idx >> 5;
    const int ks = t % KT; t /= KT;
    const int ct = t % CT;
    const int d  = t / CT;
    const int n  = (ct << 4) + (lane & 15);
    const int kk = (ks << 2) + ((lane >> 4) << 1);
    float v0 = 0.f, v1 = 0.f;
    if (n < 4 * H_DIM) {
        const float* row = Whh + ((size_t)d * 4 * H_DIM + n) * H_DIM;
        if (kk     < H_DIM) v0 = row[kk];
        if (kk + 1 < H_DIM) v1 = row[kk + 1];
    }
    outP[(size_t)idx * 2]     = v0;
    outP[(size_t)idx * 2 + 1] = v1;
}

// ---------------------------------------------------------------------------
// biLSTM cell on the WMMA pipe. Block = 64 threads = 2 waves; wave == dir.
// Each workgroup owns 16 sequence rows; per timestep it seeds 13 gate tiles
// with the precomputed input projection (C operand) and accumulates
// h(16x52) @ WhhT(52x208) via 13x13 v_wmma_f32_16x16x4_f32.
//   content mode (isNeigh=0, k=2): fwd t0<-Ga,t1<-Gb; rev t0<-Gb,t1<-Ga
//   neigh  mode (isNeigh=1):       pre[(n*k + t)*400 + d*200 + col], rev flips t
// c-state and running mean live in registers (fixed lane->(row,j) map).
// ---------------------------------------------------------------------------
__global__ __launch_bounds__(64) void lstm_cell_wmma(
    const float* __restrict__ Ga, const float* __restrict__ Gb,
    const float* __restrict__ WhhP,      // packed (2,13,13,32) v2f
    float* __restrict__ outAgg, int M, int k, int isNeigh)
{
    __shared__ float Hs[2][16 * HS_STRIDE];   // h per dir (cols 50..51 stay 0)
    __shared__ float Gs[2][16 * GS_STRIDE];   // gate preacts per dir

    const int d    = threadIdx.x >> 5;        // wave == direction
    const int lane = threadIdx.x & 31;
    const int rowBase = blockIdx.x << 4;
    const int lh   = lane & 15;
    const int kofs = (lane >> 4) << 1;
    const int rAdd = (lane >> 4) << 3;

    float* Hw = &Hs[d][0];
    float* Gw = &Gs[d][0];
    const float* Wp = WhhP + (size_t)d * CT * KT * 32 * 2;

    float cst[25], mst[25];
#pragma unroll
    for (int q = 0; q < 25; ++q) { cst[q] = 0.f; mst[q] = 0.f; }
    for (int e = lane; e < 16 * HS_STRIDE; e += 32) Hw[e] = 0.f;   // h0 = 0 (+K pad)
    __builtin_amdgcn_wave_barrier();

    for (int s = 0; s < k; ++s) {
        for (int ct = 0; ct < CT; ++ct) {
            const int colBase = ct << 4;
            const int col = colBase + lh;
            // seed C with the input-projection preactivation tile
            v8f acc;
#pragma unroll
            for (int r = 0; r < 8; ++r) {
                const int row = rowBase + rAdd + r;
                float pv = 0.f;
                if (col < 4 * H_DIM) {
                    if (isNeigh) {
                        const int tt = d ? (k - 1 - s) : s;
                        pv = Ga[((size_t)row * k + tt) * G4H + d * 200 + col];
                    } else {
                        const float* buf = ((s ^ d) & 1) ? Gb : Ga;
                        pv = buf[(size_t)row * G4H + d * 200 + col];
                    }
                }
                acc[r] = pv;
            }
            // acc += h @ WhhT tile
#pragma unroll
            for (int ks = 0; ks < KT; ++ks) {
                v2f a = *(const v2f*)&Hw[lh * HS_STRIDE + (ks << 2) + kofs];
                v2f b = *(const v2f*)&Wp[(((size_t)ct * KT + ks) * 32 + lane) * 2];
                acc = wmma_f32(a, b, acc);
            }
#pragma unroll
            for (int r = 0; r < 8; ++r)
                Gw[(rAdd + r) * GS_STRIDE + col] = acc[r];
        }
        __builtin_amdgcn_wave_barrier();

        // elementwise LSTM update: 16 rows x 50 = 800 items, 25 per lane
#pragma unroll
        for (int q = 0; q < 25; ++q) {
            const int e = lane + (q << 5);
            const int row = e / H_DIM, j = e - row * H_DIM;
            const float gi = Gw[row * GS_STRIDE + j];
            const float gf = Gw[row * GS_STRIDE + H_DIM + j];
            const float gg = Gw[row * GS_STRIDE + 2 * H_DIM + j];
            const float go = Gw[row * GS_STRIDE + 3 * H_DIM + j];
            const float cv = sigf(gf) * cst[q] + sigf(gi) * tanhf(gg);
            const float hv = sigf(go) * tanhf(cv);
            cst[q] = cv;
            mst[q] += hv;
            Hw[row * HS_STRIDE + j] = hv;
        }
        __builtin_amdgcn_wave_barrier();
    }

    const float inv = 1.0f / (float)k;
#pragma unroll
    for (int q = 0; q < 25; ++q) {
        const int e = lane + (q << 5);
        const int row = e / H_DIM, j = e - row * H_DIM;
        outAgg[(size_t)(rowBase + row) * D_EMB + d * H_DIM + j] = mst[q] * inv;
    }
    (void)M;
}

// Attention over {c,u,i,s} aggregates + padding mask.
__global__ void attention_combine(const float* __restrict__ cA, const float* __restrict__ uA,
                                  const float* __restrict__ iA, const float* __restrict__ sA,
                                  const float* __restrict__ att,     // (200,)
                                  const int* __restrict__ flat,
                                  float* __restrict__ out, int N)
{
    const int n = blockIdx.x * blockDim.x + threadIdx.x;
    if (n >= N) return;
    if (flat[n] == 0) {
        for (int j = 0; j < D_EMB; ++j) out[(size_t)n * D_EMB + j] = 0.f;
        return;
    }
    const float* X[4] = { cA, uA, iA, sA };
    float base = 0.f;
    for (int j = 0; j < D_EMB; ++j) base += att[j] * cA[(size_t)n * D_EMB + j];
    float sc[4];
    float mx = -3.4e38f;
    for (int q = 0; q < 4; ++q) {
        float s2 = 0.f;
        for (int j = 0; j < D_EMB; ++j) s2 += att[D_EMB + j] * X[q][(size_t)n * D_EMB + j];
        float v = base + s2;
        v = (v > 0.f) ? v : 0.01f * v;                    // leaky_relu, slope 0.01
        sc[q] = v; mx = fmaxf(mx, v);
    }
    float den = 0.f;
    for (int q = 0; q < 4; ++q) { sc[q] = __expf(sc[q] - mx); den += sc[q]; }
    const float inv = 1.0f / den;
    for (int j = 0; j < D_EMB; ++j) {
        float acc = 0.f;
        for (int q = 0; q < 4; ++q) acc += sc[q] * X[q][(size_t)n * D_EMB + j];
        out[(size_t)n * D_EMB + j] = acc * inv;
    }
}

// id gather: out[g] = neigh[flat[g/k]*k + g%k]
__global__ void make_ids(const int* __restrict__ flat, const int* __restrict__ neigh,
                         int k, int* __restrict__ out, int total)
{
    const int g = blockIdx.x * blockDim.x + threadIdx.x;
    if (g >= total) return;
    out[g] = neigh[(size_t)flat[g / k] * k + (g % k)];
}

// ---------------------------------------------------------------------------
extern "C" void kernel_launch(void* const* d_in, const int* in_sizes, int n_in,
                              void* d_out, int out_size, void* d_ws, size_t ws_size,
                              hipStream_t stream)
{
    const int*   flat = (const int*)  d_in[0];   // (B*T,) item ids, already flat
    const float* f1   = (const float*)d_in[1];
    const float* f3   = (const float*)d_in[2];
    const float* f5   = (const float*)d_in[3];
    const float* f8   = (const float*)d_in[4];
    const int*   nu   = (const int*)  d_in[5];
    const int*   ni   = (const int*)  d_in[6];
    const int*   ns   = (const int*)  d_in[7];
    const float* ucW  = (const float*)d_in[8],  *ucU = (const float*)d_in[9],  *ucB = (const float*)d_in[10];
    const float* icW  = (const float*)d_in[11], *icU = (const float*)d_in[12], *icB = (const float*)d_in[13];
    const float* scW  = (const float*)d_in[14], *scU = (const float*)d_in[15], *scB = (const float*)d_in[16];
    const float* unW  = (const float*)d_in[17], *unU = (const float*)d_in[18], *unB = (const float*)d_in[19];
    const float* inW  = (const float*)d_in[20], *inU = (const float*)d_in[21], *inB = (const float*)d_in[22];
    const float* att  = (const float*)d_in[23];

    const int N  = in_sizes[0];      // 12800
    const int N5 = N * 5;
    const int N2 = N * 2;

    // ---- workspace carve (bump allocator, 256B aligned) ----
    char*  ws  = (char*)d_ws;
    size_t off = 0;
    auto alloc = [&](size_t bytes) -> void* {
        void* p = ws + off;
        off += (bytes + 255) & ~(size_t)255;
        return p;
    };
    const size_t PACK_ELEMS = (size_t)2 * CT * KT * 32 * 2;   // floats per packed Whh
    int*   ids_u = (int*)  alloc((size_t)N5 * 4);
    int*   ids_i = (int*)  alloc((size_t)N5 * 4);
    int*   ids_s = (int*)  alloc((size_t)N2 * 4);
    float* ucP   = (float*)alloc(PACK_ELEMS * 4);
    float* icP   = (float*)alloc(PACK_ELEMS * 4);
    float* scP   = (float*)alloc(PACK_ELEMS * 4);
    float* unP   = (float*)alloc(PACK_ELEMS * 4);
    float* inP   = (float*)alloc(PACK_ELEMS * 4);
    float* preA  = (float*)alloc((size_t)N5 * G4H * 4);   // 102.4 MB
    float* preB  = (float*)alloc((size_t)N5 * G4H * 4);   // 102.4 MB
    float* Cbuf  = (float*)alloc((size_t)N5 * D_EMB * 4); // 25.6 MB
    float* uAgg  = (float*)alloc((size_t)N * D_EMB * 4);
    float* iAgg  = (float*)alloc((size_t)N * D_EMB * 4);
    float* sAgg  = (float*)alloc((size_t)N * D_EMB * 4);
    float* cAgg  = (float*)alloc((size_t)N * D_EMB * 4);
    (void)ws_size; (void)n_in; (void)out_size;

    const dim3 blk64(64), blk128(128), blk256(256);
    auto cdiv = [](int a, int b) { return (a + b - 1) / b; };
    const int packThreads = 2 * CT * KT * 32;

    // ---- pack all Whh matrices into WMMA B-fragment order ----
    pack_whh<<<cdiv(packThreads, 256), blk256, 0, stream>>>(ucU, ucP);
    pack_whh<<<cdiv(packThreads, 256), blk256, 0, stream>>>(icU, icP);
    pack_whh<<<cdiv(packThreads, 256), blk256, 0, stream>>>(scU, scP);
    pack_whh<<<cdiv(packThreads, 256), blk256, 0, stream>>>(unU, unP);
    pack_whh<<<cdiv(packThreads, 256), blk256, 0, stream>>>(inU, inP);

    // ---- id gathers ----
    make_ids<<<cdiv(N5, 256), blk256, 0, stream>>>(flat, nu, 5, ids_u, N5);
    make_ids<<<cdiv(N5, 256), blk256, 0, stream>>>(flat, ni, 5, ids_i, N5);
    make_ids<<<cdiv(N2, 256), blk256, 0, stream>>>(flat, ns, 2, ids_s, N2);

    // ---- user branch: content (ta==tb==f1) -> neigh (k=5) ----
    gemm_gather_wmma<<<N5 / 16, blk128, 0, stream>>>(f1, ids_u, ucW, ucB, preA, N5);
    lstm_cell_wmma<<<N5 / 16, blk64, 0, stream>>>(preA, preA, ucP, Cbuf, N5, 2, 0);
    gemm_gather_wmma<<<N5 / 16, blk128, 0, stream>>>(Cbuf, nullptr, unW, unB, preB, N5);
    lstm_cell_wmma<<<N / 16, blk64, 0, stream>>>(preB, nullptr, unP, uAgg, N, 5, 1);

    // ---- item branch: content (f3, f5) -> neigh (k=5) ----
    gemm_gather_wmma<<<N5 / 16, blk128, 0, stream>>>(f3, ids_i, icW, icB, preA, N5);
    gemm_gather_wmma<<<N5 / 16, blk128, 0, stream>>>(f5, ids_i, icW, icB, preB, N5);
    lstm_cell_wmma<<<N5 / 16, blk64, 0, stream>>>(preA, preB, icP, Cbuf, N5, 2, 0);
    gemm_gather_wmma<<<N5 / 16, blk128, 0, stream>>>(Cbuf, nullptr, inW, inB, preA, N5);
    lstm_cell_wmma<<<N / 16, blk64, 0, stream>>>(preA, nullptr, inP, iAgg, N, 5, 1);

    // ---- session branch: content (f8) -> neigh (k=2, item-neigh weights: source bug kept) ----
    gemm_gather_wmma<<<N2 / 16, blk128, 0, stream>>>(f8, ids_s, scW, scB, preA, N2);
    lstm_cell_wmma<<<N2 / 16, blk64, 0, stream>>>(preA, preA, scP, Cbuf, N2, 2, 0);
    gemm_gather_wmma<<<N2 / 16, blk128, 0, stream>>>(Cbuf, nullptr, inW, inB, preB, N2);
    lstm_cell_wmma<<<N / 16, blk64, 0, stream>>>(preB, nullptr, inP, sAgg, N, 2, 1);

    // ---- self content over flat ids (f3, f5 with item-content weights) ----
    gemm_gather_wmma<<<N / 16, blk128, 0, stream>>>(f3, flat, icW, icB, preA, N);
    gemm_gather_wmma<<<N / 16, blk128, 0, stream>>>(f5, flat, icW, icB, preB, N);
    lstm_cell_wmma<<<N / 16, blk64, 0, stream>>>(preA, preB, icP, cAgg, N, 2, 0);

    // ---- attention + padding mask ----
    attention_combine<<<cdiv(N, 256), blk256, 0, stream>>>(
        cAgg, uAgg, iAgg, sAgg, att, flat, (float*)d_out, N);
}